// Blur_13073880449610
// MI455X (gfx1250) — compile-verified
//
#include <hip/hip_runtime.h>
#include <stdint.h>

// Padded input tile in LDS: padded coords (r in [0,67), p in [0,67)).
// Element (r,p) stored at float index r*TSTRIDE + p + TSKEW.
// TSTRIDE=72 so each 64-float interior row is followed by an 8-float gap that
// the TDM pad feature (interval 64 DW, amount 8 DW) produces in hardware.
#define TSTRIDE 72
#define TROWS   67
#define TSKEW   2
#define HROWS   67     // horizontally-filtered rows (padded rows), 64 cols each

typedef uint32_t u32;
typedef uint64_t u64;
typedef u32   u32x4 __attribute__((ext_vector_type(4)));
typedef u32   u32x8 __attribute__((ext_vector_type(8)));
typedef float f32x4 __attribute__((ext_vector_type(4)));

__global__ __launch_bounds__(256) void blur_sep_tdm_kernel(
        const float* __restrict__ x,
        const float* __restrict__ kern,
        float* __restrict__ out)
{
    __shared__ __align__(16) float tile[TSTRIDE * TROWS];  // 19296 B (1206 x f32x4)
    __shared__ __align__(16) float htmp[HROWS * 64];       // 17152 B

    const int tid = threadIdx.x;
    const float* src = x   + (size_t)blockIdx.x * 4096;    // one (b,c) plane
    float*       dst = out + (size_t)blockIdx.x * 4096;

    // ---- 1) zero the padded tile with b128 LDS stores (halo must be zero;
    //         the DMA overwrites the interior, ordered by the barrier).
    //         1206 quads = 4*256 uniform + 182 tail (no loop-carried exec mask).
    {
        f32x4* t4 = reinterpret_cast<f32x4*>(tile);
        const f32x4 z = {0.f, 0.f, 0.f, 0.f};
#pragma unroll
        for (int k = 0; k < 4; ++k) t4[tid + k * 256] = z;
        if (tid < (TSTRIDE * TROWS) / 4 - 1024) t4[tid + 1024] = z;
    }
    __syncthreads();

    // ---- 2) Tensor Data Mover: one descriptor moves the 64x64 f32 tile,
    //         pad_enable inserts the 8-DWORD LDS gap after every 64-DWORD row.
    if (tid < 32) {   // wave 0 only (TDM ignores EXEC; one DMA per block)
        const u32 lds_interior =
            (u32)(uintptr_t)(&tile[0]) + (u32)((2 * TSTRIDE + 2 + TSKEW) * 4);
        const u64 ga = (u64)(uintptr_t)src;

        u32x4 g0;
        g0[0] = 1u;                                   // count=1 (valid), user D#
        g0[1] = lds_interior;                         // lds_addr (bytes)
        g0[2] = (u32)ga;                              // global_addr[31:0]
        g0[3] = (u32)((ga >> 32) & 0x01FFFFFFu)       // global_addr[56:32]
              | (2u << 30);                           // type=2 ("image")

        u32x8 g1;
        g1[0] = (2u << 16)      // data_size = 4 bytes
              | (1u << 20)      // pad_enable
              | (5u << 22)      // pad_interval: 64 DWORDs
              | (7u << 25);     // pad_amount:   8 DWORDs
        g1[1] = 64u << 16;      // tensor_dim0[15:0] = 64 (cols)
        g1[2] = 64u << 16;      // tensor_dim1[15:0] = 64 (rows)
        g1[3] = 64u << 16;      // tile_dim0 = 64
        g1[4] = 64u;            // tile_dim1 = 64 (tile_dim2 = 0)
        g1[5] = 64u;            // tensor_dim0_stride = 64 elements
        g1[6] = 0x1000u << 16;  // tensor_dim1_stride[15:0] = 4096 (unused, 2D)
        g1[7] = 0u;

        asm volatile("tensor_load_to_lds %0, %1" :: "s"(g0), "s"(g1) : "memory");
        __builtin_amdgcn_s_wait_tensorcnt(0);
    }
    __syncthreads();

    // ---- 3) separable factors from the provided 4x4 kernel (rank-1).
    // Reference correlates with flipped kernel: wf[i][j] = K[3-i][3-j]
    //   = v[i]*h[j],  h[j] = K[0][3-j],  v[i] = K[3-i][0]/K[0][0]   (exact).
    const float h0 = kern[3], h1 = kern[2], h2 = kern[1], h3 = kern[0];
    const float inv = 1.0f / kern[0];
    const float v0 = kern[12] * inv;
    const float v1 = kern[8]  * inv;
    const float v2 = kern[4]  * inv;
    const float v3 = kern[0]  * inv;

    // ---- 4) horizontal pass, 4 outputs/iter via b128 LDS loads:
    //         htmp[r][4q+t] = sum_j tile-row[r][4q+t+j+TSKEW] * h[j]
    //         1072 quads = 4*256 uniform (rows 0..63) + 48 tail (rows 64..66).
    {
        const f32x4* t4 = reinterpret_cast<const f32x4*>(tile);
        f32x4*       h4 = reinterpret_cast<f32x4*>(htmp);
#pragma unroll
        for (int k = 0; k < 5; ++k) {
            const int i = tid + k * 256;
            if (k == 4 && tid >= HROWS * 16 - 1024) break;   // tail predicate
            const int r  = i >> 4;
            const int q  = i & 15;              // quad index, xq = 4*q
            const int b4 = (r * TSTRIDE) / 4 + q;
            const f32x4 A  = t4[b4];            // row floats [4q   .. 4q+3]
            const f32x4 B  = t4[b4 + 1];        // row floats [4q+4 .. 4q+7]
            const float e8 = tile[r * TSTRIDE + 4 * q + 8];
            f32x4 o;
            o[0] = A[2] * h0 + A[3] * h1 + B[0] * h2 + B[1] * h3;
            o[1] = A[3] * h0 + B[0] * h1 + B[1] * h2 + B[2] * h3;
            o[2] = B[0] * h0 + B[1] * h1 + B[2] * h2 + B[3] * h3;
            o[3] = B[1] * h0 + B[2] * h1 + B[3] * h2 + e8   * h3;
            h4[(r << 4) + q] = o;
        }
    }
    __syncthreads();

    // ---- 5) vertical pass + coalesced non-temporal b128 stores:
    //         out[y][x] = sum_i htmp[y+i][x] * v[i]   (1024 quads = 4*256 exact)
    {
        const f32x4* h4 = reinterpret_cast<const f32x4*>(htmp);
#pragma unroll
        for (int j = 0; j < 4; ++j) {
            const int q  = tid + j * 256;      // quad id 0..1023
            const int y  = q >> 4;             // output row 0..63
            const int qc = q & 15;             // quad col, x = 4*qc
            const f32x4 a = h4[((y + 0) << 4) + qc];
            const f32x4 b = h4[((y + 1) << 4) + qc];
            const f32x4 c = h4[((y + 2) << 4) + qc];
            const f32x4 d = h4[((y + 3) << 4) + qc];
            f32x4 o;
            o[0] = a[0] * v0 + b[0] * v1 + c[0] * v2 + d[0] * v3;
            o[1] = a[1] * v0 + b[1] * v1 + c[1] * v2 + d[1] * v3;
            o[2] = a[2] * v0 + b[2] * v1 + c[2] * v2 + d[2] * v3;
            o[3] = a[3] * v0 + b[3] * v1 + c[3] * v2 + d[3] * v3;
            __builtin_nontemporal_store(
                o, reinterpret_cast<f32x4*>(&dst[(y << 6) + (qc << 2)]));
        }
    }
}

extern "C" void kernel_launch(void* const* d_in, const int* in_sizes, int n_in,
                              void* d_out, int out_size, void* d_ws, size_t ws_size,
                              hipStream_t stream) {
    (void)in_sizes; (void)n_in; (void)d_ws; (void)ws_size; (void)out_size;
    const float* x    = (const float*)d_in[0];   // (32,256,64,64) f32
    const float* kern = (const float*)d_in[1];   // (4,4) f32
    float* out        = (float*)d_out;           // (32,256,64,64) f32

    const int nplanes = 32 * 256;                // one block per (b,c) plane
    blur_sep_tdm_kernel<<<dim3(nplanes), dim3(256), 0, stream>>>(x, kern, out);
}